// MBCA_3375844294869
// MI455X (gfx1250) — compile-verified
//
#include <hip/hip_runtime.h>
#include <math.h>

#define HIDN 512
#define NHEAD 8
#define DKH 64
#define LDRUG 512
#define LPROT 1024
#define BATCH 16

typedef __attribute__((ext_vector_type(2))) float v2f;
typedef __attribute__((ext_vector_type(8))) float v8f;

// One K=4 step of a 16x16 fp32 WMMA tile: D = A(16x4) * B(4x16) + C
// Lane mapping (ISA 7.12.2, 32-bit):
//   A: lane%16 = M row, half=lane/16 selects K pair {2h,2h+1} in vgprs {0,1}
//   B: lane%16 = N col, half selects K pair {2h,2h+1}
//   C/D: vgpr v -> row v+8*half, lane%16 -> col
__device__ __forceinline__ v8f wmma_f32x4(v8f C, float a0, float a1, float b0, float b1) {
  v2f A = {a0, a1};
  v2f B = {b0, b1};
  return __builtin_amdgcn_wmma_f32_16x16x4_f32(false, A, false, B, (short)0, C, false, false);
}

// Reference PE quirk: pe sliced by batch index -> per-batch constant vector over H.
__device__ __forceinline__ float pe_val(int b, int d) {
  const float kdiv = 9.210340371976184f / 512.0f; // ln(10000)/H
  float div = __expf(-(float)(d & ~1) * kdiv);
  float ang = (float)b * div;
  return (d & 1) ? __cosf(ang) : __sinf(ang);
}

// OUT[m, n] = sum_c (X[m,c] + pe(b,c)) * W[n,c] + bias[n]
// grid: (rows/128, HIDN/64), block 256 (8 waves, each: 16 rows x 64 cols)
__global__ void __launch_bounds__(256) mbca_proj_kernel(
    const float* __restrict__ X, const float* __restrict__ W,
    const float* __restrict__ bias, float* __restrict__ out, int L) {
  __shared__ float spe[HIDN];
  const int lane = threadIdx.x & 31;
  const int wave = threadIdx.x >> 5;
  const int half = lane >> 4;
  const int r = lane & 15;
  const int bidx = (blockIdx.x * 128) / L;  // all 8 waves share one batch idx
  for (int i = threadIdx.x; i < HIDN; i += 256) spe[i] = pe_val(bidx, i);
  __syncthreads();

  const int m0 = (blockIdx.x * 8 + wave) * 16;
  const int n0 = blockIdx.y * 64;
  const float* xrow = X + (size_t)(m0 + r) * HIDN;
  const float* w0 = W + (size_t)(n0 + r) * HIDN;
  const float* w1 = w0 + 16 * HIDN;
  const float* w2 = w0 + 32 * HIDN;
  const float* w3 = w0 + 48 * HIDN;
  v8f acc[4];
  acc[0] = (v8f){}; acc[1] = (v8f){}; acc[2] = (v8f){}; acc[3] = (v8f){};
  for (int c = 0; c < HIDN; c += 4) {
    int ca = c + 2 * half;
    float a0 = xrow[ca] + spe[ca];
    float a1 = xrow[ca + 1] + spe[ca + 1];
    acc[0] = wmma_f32x4(acc[0], a0, a1, w0[ca], w0[ca + 1]);
    acc[1] = wmma_f32x4(acc[1], a0, a1, w1[ca], w1[ca + 1]);
    acc[2] = wmma_f32x4(acc[2], a0, a1, w2[ca], w2[ca + 1]);
    acc[3] = wmma_f32x4(acc[3], a0, a1, w3[ca], w3[ca + 1]);
  }
#pragma unroll
  for (int t = 0; t < 4; ++t) {
    int n = n0 + t * 16 + r;
    float bv = bias[n];
#pragma unroll
    for (int v = 0; v < 8; ++v) {
      int row = m0 + v + 8 * half;
      out[(size_t)row * HIDN + n] = acc[t][v] + bv;
    }
  }
}

// Row-wise (max, sumexp) of S = A * B^T * 0.125 over the head slice.
// A: [B*La, HIDN] rows (head h at col h*64), B: [B*Lb, HIDN].
// grid (La/128, NHEAD, BATCH), block 256; wave owns 16 rows, streams all Lb cols.
__global__ void __launch_bounds__(256) mbca_rowstats_kernel(
    const float* __restrict__ A, const float* __restrict__ Bm,
    float* __restrict__ outM, float* __restrict__ outZ, int La, int Lb) {
  const int lane = threadIdx.x & 31;
  const int wave = threadIdx.x >> 5;
  const int half = lane >> 4;
  const int r = lane & 15;
  const int h = blockIdx.y, b = blockIdx.z;
  const int row0 = (blockIdx.x * 8 + wave) * 16;

  const float* arow = A + ((size_t)b * La + row0 + r) * HIDN + h * DKH;
  float af[32];
#pragma unroll
  for (int s = 0; s < 16; ++s) {
    int ca = s * 4 + 2 * half;
    af[2 * s] = arow[ca];
    af[2 * s + 1] = arow[ca + 1];
  }
  float runM[8], runZ[8];
#pragma unroll
  for (int v = 0; v < 8; ++v) { runM[v] = -1e30f; runZ[v] = 0.f; }

  for (int k0 = 0; k0 < Lb; k0 += 16) {
    const float* brow = Bm + ((size_t)b * Lb + k0 + r) * HIDN + h * DKH;
    v8f C = (v8f){};
#pragma unroll
    for (int s = 0; s < 16; ++s) {
      int ca = s * 4 + 2 * half;
      C = wmma_f32x4(C, af[2 * s], af[2 * s + 1], brow[ca], brow[ca + 1]);
    }
#pragma unroll
    for (int v = 0; v < 8; ++v) {
      float sv = C[v] * 0.125f;
      float nm = fmaxf(runM[v], sv);
      runZ[v] = runZ[v] * __expf(runM[v] - nm) + __expf(sv - nm);
      runM[v] = nm;
    }
  }
  // merge 16 lanes within each half (each lane held a distinct column subset)
#pragma unroll
  for (int off = 1; off < 16; off <<= 1) {
#pragma unroll
    for (int v = 0; v < 8; ++v) {
      float oM = __shfl_xor(runM[v], off, 32);
      float oZ = __shfl_xor(runZ[v], off, 32);
      float nm = fmaxf(runM[v], oM);
      runZ[v] = runZ[v] * __expf(runM[v] - nm) + oZ * __expf(oM - nm);
      runM[v] = nm;
    }
  }
  float wm = 0.f, wz = 0.f;
#pragma unroll
  for (int v = 0; v < 8; ++v)
    if ((lane & 7) == v) { wm = runM[v]; wz = runZ[v]; }
  if (r < 8) {
    size_t o = ((size_t)b * NHEAD + h) * La + row0 + r + 8 * half;
    outM[o] = wm;
    outZ[o] = wz;
  }
}

// outS[b,h,row] = sum_col exp(S[row,col] - cM[col]) / cZ[col]
__global__ void __launch_bounds__(256) mbca_crosssum_kernel(
    const float* __restrict__ A, const float* __restrict__ Bm,
    const float* __restrict__ cM, const float* __restrict__ cZ,
    float* __restrict__ outS, int La, int Lb) {
  const int lane = threadIdx.x & 31;
  const int wave = threadIdx.x >> 5;
  const int half = lane >> 4;
  const int r = lane & 15;
  const int h = blockIdx.y, b = blockIdx.z;
  const int row0 = (blockIdx.x * 8 + wave) * 16;

  const float* arow = A + ((size_t)b * La + row0 + r) * HIDN + h * DKH;
  float af[32];
#pragma unroll
  for (int s = 0; s < 16; ++s) {
    int ca = s * 4 + 2 * half;
    af[2 * s] = arow[ca];
    af[2 * s + 1] = arow[ca + 1];
  }
  float acc[8];
#pragma unroll
  for (int v = 0; v < 8; ++v) acc[v] = 0.f;

  for (int k0 = 0; k0 < Lb; k0 += 16) {
    size_t co = ((size_t)b * NHEAD + h) * Lb + k0 + r;
    float mc = cM[co];
    float iz = 1.0f / cZ[co];
    const float* brow = Bm + ((size_t)b * Lb + k0 + r) * HIDN + h * DKH;
    v8f C = (v8f){};
#pragma unroll
    for (int s = 0; s < 16; ++s) {
      int ca = s * 4 + 2 * half;
      C = wmma_f32x4(C, af[2 * s], af[2 * s + 1], brow[ca], brow[ca + 1]);
    }
#pragma unroll
    for (int v = 0; v < 8; ++v)
      acc[v] += __expf(C[v] * 0.125f - mc) * iz;
  }
#pragma unroll
  for (int off = 1; off < 16; off <<= 1)
#pragma unroll
    for (int v = 0; v < 8; ++v)
      acc[v] += __shfl_xor(acc[v], off, 32);
  float wv = 0.f;
#pragma unroll
  for (int v = 0; v < 8; ++v)
    if ((lane & 7) == v) wv = acc[v];
  if (r < 8)
    outS[((size_t)b * NHEAD + h) * La + row0 + r + 8 * half] = wv;
}

// attn_out[b,q,j] = sigmoid(sum_h (inv + inv*S[b,h,q]) * Wg[j,h] + bg[j])
// gated = (X+pe)*(1+attn); partial max over 128-q chunk -> pmax
__global__ void __launch_bounds__(512) mbca_gate_kernel(
    const float* __restrict__ X, const float* __restrict__ S,
    const float* __restrict__ Wg, const float* __restrict__ bg,
    float* __restrict__ attn_out, float* __restrict__ pmax,
    int L, float inv) {
  const int b = blockIdx.x;
  const int qc = blockIdx.y;
  const int j = threadIdx.x;
  float w[NHEAD];
#pragma unroll
  for (int hh = 0; hh < NHEAD; ++hh) w[hh] = Wg[j * NHEAD + hh];
  const float bj = bg[j];
  const float pe = pe_val(b, j);
  float mx = -1e30f;
  const int q0 = qc * 128;
  for (int q = q0; q < q0 + 128; ++q) {
    float v = bj;
#pragma unroll
    for (int hh = 0; hh < NHEAD; ++hh) {
      float mr = fmaf(inv, S[((size_t)b * NHEAD + hh) * L + q], inv);
      v = fmaf(mr, w[hh], v);
    }
    float a = 1.0f / (1.0f + __expf(-v));
    size_t xo = ((size_t)b * L + q) * HIDN + j;
    attn_out[xo] = a;
    float x = X[xo] + pe;
    float g = x + x * a;
    mx = fmaxf(mx, g);
  }
  pmax[((size_t)b * gridDim.y + qc) * HIDN + j] = mx;
}

__global__ void mbca_pairmax_kernel(const float* __restrict__ pmax,
                                    float* __restrict__ pair, int nch, int off) {
  int idx = blockIdx.x * blockDim.x + threadIdx.x;  // b*512 + j
  int b = idx >> 9, j = idx & 511;
  float mx = -1e30f;
  for (int c = 0; c < nch; ++c)
    mx = fmaxf(mx, pmax[((size_t)b * nch + c) * HIDN + j]);
  pair[b * (2 * HIDN) + off + j] = mx;
}

extern "C" void kernel_launch(void* const* d_in, const int* in_sizes, int n_in,
                              void* d_out, int out_size, void* d_ws, size_t ws_size,
                              hipStream_t stream) {
  (void)in_sizes; (void)n_in; (void)out_size; (void)ws_size;
  const float* drug = (const float*)d_in[0];
  const float* prot = (const float*)d_in[1];
  const float* Wq = (const float*)d_in[2];
  const float* bq = (const float*)d_in[3];
  const float* Wk = (const float*)d_in[4];
  const float* bk = (const float*)d_in[5];
  const float* Wdp = (const float*)d_in[6];
  const float* bdp = (const float*)d_in[7];
  const float* Wpd = (const float*)d_in[8];
  const float* bpd = (const float*)d_in[9];

  float* out = (float*)d_out;
  float* pair = out;                                   // [16, 1024]
  float* drug_attn = out + BATCH * 2 * HIDN;           // [16, 512, 512]
  float* prot_attn = drug_attn + (size_t)BATCH * LDRUG * HIDN;  // [16,1024,512]

  float* ws = (float*)d_ws;
  float* Q = ws;    ws += (size_t)BATCH * LDRUG * HIDN;   // 4M
  float* K = ws;    ws += (size_t)BATCH * LPROT * HIDN;   // 8M
  float* rowM = ws; ws += BATCH * NHEAD * LDRUG;
  float* rowZ = ws; ws += BATCH * NHEAD * LDRUG;
  float* colM = ws; ws += BATCH * NHEAD * LPROT;
  float* colZ = ws; ws += BATCH * NHEAD * LPROT;
  float* rowS = ws; ws += BATCH * NHEAD * LDRUG;
  float* colS = ws; ws += BATCH * NHEAD * LPROT;
  float* pmax = ws; ws += (size_t)BATCH * 8 * HIDN;

  dim3 blk(256);
  // Projections (PE fused)
  mbca_proj_kernel<<<dim3(BATCH * LDRUG / 128, HIDN / 64), blk, 0, stream>>>(drug, Wq, bq, Q, LDRUG);
  mbca_proj_kernel<<<dim3(BATCH * LPROT / 128, HIDN / 64), blk, 0, stream>>>(prot, Wk, bk, K, LPROT);
  // Row softmax stats of S; col stats via transposed roles (S^T = K*Q^T scaled)
  mbca_rowstats_kernel<<<dim3(LDRUG / 128, NHEAD, BATCH), blk, 0, stream>>>(Q, K, rowM, rowZ, LDRUG, LPROT);
  mbca_rowstats_kernel<<<dim3(LPROT / 128, NHEAD, BATCH), blk, 0, stream>>>(K, Q, colM, colZ, LPROT, LDRUG);
  // rowS[q] = sum_k softmax_col;  colS[k] = sum_q softmax_row
  mbca_crosssum_kernel<<<dim3(LDRUG / 128, NHEAD, BATCH), blk, 0, stream>>>(Q, K, colM, colZ, rowS, LDRUG, LPROT);
  mbca_crosssum_kernel<<<dim3(LPROT / 128, NHEAD, BATCH), blk, 0, stream>>>(K, Q, rowM, rowZ, colS, LPROT, LDRUG);
  // Gating + residual + seq max-pool
  mbca_gate_kernel<<<dim3(BATCH, LDRUG / 128), dim3(512), 0, stream>>>(
      drug, rowS, Wdp, bdp, drug_attn, pmax, LDRUG, 0.5f / (float)LPROT);
  mbca_pairmax_kernel<<<dim3(BATCH * HIDN / 256), dim3(256), 0, stream>>>(pmax, pair, LDRUG / 128, 0);
  mbca_gate_kernel<<<dim3(BATCH, LPROT / 128), dim3(512), 0, stream>>>(
      prot, colS, Wpd, bpd, prot_attn, pmax, LPROT, 0.5f / (float)LDRUG);
  mbca_pairmax_kernel<<<dim3(BATCH * HIDN / 256), dim3(256), 0, stream>>>(pmax, pair, LPROT / 128, HIDN);
}